// RetinaFacePipeline_44006234915160
// MI455X (gfx1250) — compile-verified
//
#include <hip/hip_runtime.h>
#include <cfloat>
#include <climits>
#include <cstdint>

// ---- Problem constants (from the reference) -------------------------------
#define IM_Hc 1024.0f
#define IM_Wc 1024.0f
#define VAR0c 0.1f
#define B_BATCH 128
#define N_ELEM  43008          // = 21 * 2048
#define BPB     21             // blocks per batch
#define CHUNK   2048           // score elements (n-values) per block
#define TPB     256            // 8 waves of 32
#define NWAVES  (TPB / 32)
#define ITERS   (CHUNK / (2 * TPB))   // 4 float4 loads per thread

typedef unsigned long long u64;
typedef unsigned int       u32;

// Monotone map: float total order -> u32 total order (handles any sign).
__device__ __forceinline__ u32 f32_order(float f) {
    u32 b = __float_as_uint(f);
    return b ^ ((u32)(((int)b) >> 31) | 0x80000000u);
}

// Sortable key: high 32 = ordered value bits, low 32 = ~n.
// max(key) == max value, ties -> larger ~n == smaller n (first occurrence).
__device__ __forceinline__ u64 mk_key(float f, int n) {
    return ((u64)f32_order(f) << 32) | (u32)(~n);
}

__device__ __forceinline__ u64 kmax(u64 a, u64 b) { return a > b ? a : b; }

// ---- Kernel 1: streaming per-block argmax partials ------------------------
// conf layout: (B, N, 2) f32; score = conf[b, n, 1].
// One float4 covers pairs (n, n+1): scores at .y and .w.
__global__ __launch_bounds__(TPB) void rf_argmax_partial(
    const float* __restrict__ conf,
    u64* __restrict__ pkey)
{
    const int b     = blockIdx.x / BPB;
    const int blk   = blockIdx.x % BPB;
    const int baseN = blk * CHUNK;

    const float4* __restrict__ p =
        reinterpret_cast<const float4*>(conf + ((size_t)b * N_ELEM + baseN) * 2);

    // Issue all loads first (one clause, one wait), then combine.
    float4 f[ITERS];
    #pragma unroll
    for (int it = 0; it < ITERS; ++it)
        f[it] = p[it * TPB + threadIdx.x];

    u64 k = 0;  // any real key has bit63 set (score >= 0), so 0 is identity
    #pragma unroll
    for (int it = 0; it < ITERS; ++it) {
        const int n0 = baseN + 2 * (it * TPB + (int)threadIdx.x);
        k = kmax(k, mk_key(f[it].y, n0));
        k = kmax(k, mk_key(f[it].w, n0 + 1));
    }

    // wave32 shuffle reduction (branchless u64 max)
    #pragma unroll
    for (int off = 16; off > 0; off >>= 1)
        k = kmax(k, __shfl_down(k, off, 32));

    // cross-wave combine via LDS (8 waves)
    __shared__ u64 sk[NWAVES];
    const int lane = threadIdx.x & 31;
    const int wave = threadIdx.x >> 5;
    if (lane == 0) sk[wave] = k;
    __syncthreads();

    if (wave == 0) {
        k = (lane < NWAVES) ? sk[lane] : 0ull;
        #pragma unroll
        for (int off = NWAVES / 2; off > 0; off >>= 1)
            k = kmax(k, __shfl_down(k, off, 32));
        if (lane == 0) pkey[blockIdx.x] = k;
    }
}

// ---- Kernel 2: final per-batch reduce + landmark decode -------------------
// landms layout: (B, N, 10) f32; priors: (N, 4) f32; out: (B, 10) f32.
__global__ __launch_bounds__(32) void rf_finalize(
    const u64*   __restrict__ pkey,
    const float* __restrict__ landms,
    const float* __restrict__ priors,
    float*       __restrict__ out)
{
    const int b    = blockIdx.x;
    const int lane = threadIdx.x;

    u64 k = (lane < BPB) ? pkey[b * BPB + lane] : 0ull;
    #pragma unroll
    for (int off = 16; off > 0; off >>= 1)
        k = kmax(k, __shfl_down(k, off, 32));
    k = __shfl(k, 0, 32);                        // broadcast winning key

    const int idx = (int)(~(u32)k);              // recover first-max index

    if (lane < 10) {
        const int   j     = lane;
        const int   c     = j & 1;               // 0 = x-coord, 1 = y-coord
        const float pxy   = priors[(size_t)idx * 4 + c];
        const float pwh   = priors[(size_t)idx * 4 + 2 + c];
        const float lm    = landms[((size_t)b * N_ELEM + (size_t)idx) * 10 + j];
        // scale1 = tile([IM_W, IM_H]); norm = tile([IM_H, IM_W]); both 1024.
        const float scale = c ? IM_Hc : IM_Wc;
        const float norm  = c ? IM_Wc : IM_Hc;
        out[b * 10 + j] = (pxy + lm * VAR0c * pwh) * scale / norm;
    }
}

// ---- Launch ---------------------------------------------------------------
extern "C" void kernel_launch(void* const* d_in, const int* in_sizes, int n_in,
                              void* d_out, int out_size, void* d_ws, size_t ws_size,
                              hipStream_t stream) {
    // Inputs in setup_inputs() order: loc (unused), conf, landms, priors.
    const float* conf   = (const float*)d_in[1];
    const float* landms = (const float*)d_in[2];
    const float* priors = (const float*)d_in[3];
    float*       out    = (float*)d_out;

    // Workspace: B*BPB partial u64 keys = 2688 * 8 B = 21504 B.
    u64* pkey = (u64*)d_ws;

    rf_argmax_partial<<<B_BATCH * BPB, TPB, 0, stream>>>(conf, pkey);
    rf_finalize<<<B_BATCH, 32, 0, stream>>>(pkey, landms, priors, out);
}